// PartTokenGNNModel_15839839387999
// MI455X (gfx1250) — compile-verified
//
#include <hip/hip_runtime.h>
#include <hip/hip_bf16.h>
#include <math.h>

// ---------------- problem constants ----------------
#define NB    64          // batch (graphs)
#define NPG   512         // nodes per graph
#define NN    (NB*NPG)    // 32768 nodes
#define DEG   16
#define EE    (NN*DEG)    // 524288 edges
#define DD    128         // feature dim
#define HH    4           // heads
#define KK    5           // clusters
#define NHID  50
#define CBN   512         // codebook entries

typedef __attribute__((ext_vector_type(2))) float v2f;
typedef __attribute__((ext_vector_type(8))) float v8f;

enum { ACT_NONE = 0, ACT_RELU = 1, ACT_SIG = 2 };

// =====================================================================
// Fused GEMM on the CDNA5 matrix pipe:  C = act((A@B + bias)*g + be)
// A: [M,K] row-major, B: [K,N] row-major, C: [M,N]; N,K compile-time so
// all strides fold into global_load immediate offsets (no 64-bit index
// math in the hot loop).  One wave owns a 16 x (16*NT) strip via
// V_WMMA_F32_16X16X4_F32.  blockDim = 128 (4 waves); launches guarantee
// N == 4*NT*16 so EXEC is all-ones at every WMMA (ISA requirement).
//
// VGPR layouts per CDNA5 ISA 7.12.2:
//   A 16x4 : lane l holds A[l%16][2*(l>>4)+j]      (j = vector elem 0..1)
//   B 4x16 : lane l holds B[2*(l>>4)+j][l%16]
//   D 16x16: lane l, reg r holds D[r + 8*(l>>4)][l%16]
// =====================================================================
template <int NT, int N, int K>
__global__ __launch_bounds__(128) void wmma_gemm_f32(
    const float* __restrict__ A, const float* __restrict__ B,
    const float* __restrict__ bias, const float* __restrict__ gamma,
    const float* __restrict__ beta, float* __restrict__ C,
    int act, float bnscale)
{
    const int lane  = threadIdx.x & 31;
    const int wave  = threadIdx.x >> 5;
    const int m0    = blockIdx.x * 16;
    const int mrow  = lane & 15;
    const int khalf = lane >> 4;           // 0 or 1
    const int nbase = wave * (NT * 16);

    v8f acc[NT];
#pragma unroll
    for (int j = 0; j < NT; ++j) acc[j] = v8f{0.f,0.f,0.f,0.f,0.f,0.f,0.f,0.f};

    // constant-stride bases: per-iteration offsets are compile-time imms
    const float* Ap = A + (m0 + mrow) * K + 2 * khalf;
    const float* Bp = B + (2 * khalf) * N + nbase + mrow;

#pragma unroll 8
    for (int k0 = 0; k0 < K; k0 += 4) {
        v2f a = *(const v2f*)(Ap + k0);
#pragma unroll
        for (int j = 0; j < NT; ++j) {
            v2f b;
            b.x = Bp[k0 * N + j * 16];
            b.y = Bp[k0 * N + j * 16 + N];
            acc[j] = __builtin_amdgcn_wmma_f32_16x16x4_f32(
                false, a, false, b, (short)0, acc[j], false, false);
        }
    }

#pragma unroll
    for (int j = 0; j < NT; ++j) {
        const int n  = nbase + j * 16 + mrow;
        const float bi = bias  ? bias[n]             : 0.0f;
        const float ga = gamma ? gamma[n] * bnscale  : 1.0f;
        const float be = beta  ? beta[n]             : 0.0f;
        float* Cp = C + m0 * N + n;
#pragma unroll
        for (int r = 0; r < 8; ++r) {
            float v = acc[j][r] + bi;
            if (gamma) v = v * ga + be;
            if (act == ACT_RELU)      v = fmaxf(v, 0.0f);
            else if (act == ACT_SIG)  v = 1.0f / (1.0f + __expf(-v));
            Cp[(r + 8 * khalf) * N] = v;
        }
    }
}

// ---------------- embedding lookup: h[i,:] = emb[x[i],:] ----------------
__global__ __launch_bounds__(256) void embed_kernel(
    const int* __restrict__ x, const float* __restrict__ emb,
    float* __restrict__ h)
{
    int idx = blockIdx.x * 256 + threadIdx.x;       // NN*DD threads
    int i = idx >> 7, d = idx & 127;
    h[(size_t)i * DD + d] = emb[(size_t)x[i] * DD + d];
}

// ---------------- edge scatter-add: agg[dst] += h[src] ------------------
// 8 threads per edge, 16 floats each; fp32 atomics resolve in L2.
__global__ __launch_bounds__(256) void edge_scatter(
    const int* __restrict__ src, const int* __restrict__ dst,
    const float* __restrict__ h, float* __restrict__ agg)
{
    int idx = blockIdx.x * 256 + threadIdx.x;       // EE*8 threads
    int e = idx >> 3, c = idx & 7;
    int s = src[e], d = dst[e];
    const float4* hp = (const float4*)(h + (size_t)s * DD + c * 16);
    float* ap = agg + (size_t)d * DD + c * 16;
#pragma unroll
    for (int q = 0; q < 4; ++q) {
        float4 v = hp[q];
        __hip_atomic_fetch_add(ap + q*4 + 0, v.x, __ATOMIC_RELAXED, __HIP_MEMORY_SCOPE_AGENT);
        __hip_atomic_fetch_add(ap + q*4 + 1, v.y, __ATOMIC_RELAXED, __HIP_MEMORY_SCOPE_AGENT);
        __hip_atomic_fetch_add(ap + q*4 + 2, v.z, __ATOMIC_RELAXED, __HIP_MEMORY_SCOPE_AGENT);
        __hip_atomic_fetch_add(ap + q*4 + 3, v.w, __ATOMIC_RELAXED, __HIP_MEMORY_SCOPE_AGENT);
    }
}

// ---------------- partitioner MLP + softmax (per node) ------------------
__global__ __launch_bounds__(256) void partition_softmax(
    const float* __restrict__ h, const float* __restrict__ W1,
    const float* __restrict__ b1, const float* __restrict__ W2,
    const float* __restrict__ b2, float* __restrict__ S)
{
    int n = blockIdx.x * 256 + threadIdx.x;
    if (n >= NN) return;
    const float* row = h + (size_t)n * DD;

    float hid[NHID];
#pragma unroll
    for (int j = 0; j < NHID; ++j) hid[j] = b1[j];
    for (int d = 0; d < DD; ++d) {
        float xv = row[d];
        const float* w = W1 + d * NHID;
#pragma unroll
        for (int j = 0; j < NHID; ++j) hid[j] = fmaf(xv, w[j], hid[j]);
    }
    float cl[KK];
#pragma unroll
    for (int k = 0; k < KK; ++k) cl[k] = b2[k];
#pragma unroll
    for (int j = 0; j < NHID; ++j) {
        float v = fmaxf(hid[j], 0.0f);
#pragma unroll
        for (int k = 0; k < KK; ++k) cl[k] = fmaf(v, W2[j * KK + k], cl[k]);
    }
    float mx = cl[0];
#pragma unroll
    for (int k = 1; k < KK; ++k) mx = fmaxf(mx, cl[k]);
    float s = 0.0f;
#pragma unroll
    for (int k = 0; k < KK; ++k) { cl[k] = __expf(cl[k] - mx); s += cl[k]; }
    float inv = 1.0f / s;
#pragma unroll
    for (int k = 0; k < KK; ++k) S[(size_t)n * KK + k] = cl[k] * inv;
}

// ------------- cf[b,k,:] = (S^T @ dense) / (colsum(S)+eps) --------------
__global__ __launch_bounds__(256) void cluster_feat(
    const float* __restrict__ h, const float* __restrict__ S,
    float* __restrict__ cf)
{
    int b = blockIdx.x, t = threadIdx.x;
    __shared__ float ssum[KK];
    if (t < KK) {
        float s = 0.0f;
        const float* Sp = S + (size_t)b * NPG * KK + t;
        for (int n = 0; n < NPG; ++n) s += Sp[n * KK];
        ssum[t] = s + 1e-6f;
    }
    __syncthreads();
    for (int oi = t; oi < KK * DD; oi += 256) {
        int k = oi / DD, d = oi % DD;
        float acc = 0.0f;
        const float* hp = h + (size_t)b * NPG * DD + d;
        const float* Sp = S + (size_t)b * NPG * KK + k;
        for (int n = 0; n < NPG; ++n) acc = fmaf(Sp[n * KK], hp[n * DD], acc);
        cf[(size_t)b * KK * DD + oi] = acc / ssum[k];
    }
}

// ------------- VQ: zq[row] = codebook[argmin ||cf[row]-c||^2] -----------
__global__ __launch_bounds__(128) void vq_nearest(
    const float* __restrict__ cf, const float* __restrict__ cb,
    float* __restrict__ zq)
{
    int row = blockIdx.x, t = threadIdx.x;
    const float* xr = cf + (size_t)row * DD;
    float best = 3.4e38f; int bi = CBN;
#pragma unroll
    for (int i = 0; i < CBN / 128; ++i) {
        int e = t + i * 128;
        const float* c = cb + (size_t)e * DD;
        float dot = 0.0f, cn = 0.0f;
        for (int d = 0; d < DD; ++d) {
            float cv = c[d];
            dot = fmaf(cv, xr[d], dot);
            cn  = fmaf(cv, cv, cn);
        }
        float sc = cn - 2.0f * dot;     // |x|^2 constant, dropped
        if (sc < best || (sc == best && e < bi)) { best = sc; bi = e; }
    }
    __shared__ float sv[128]; __shared__ int si[128];
    sv[t] = best; si[t] = bi;
    __syncthreads();
    for (int off = 64; off > 0; off >>= 1) {
        if (t < off) {
            if (sv[t+off] < sv[t] || (sv[t+off] == sv[t] && si[t+off] < si[t])) {
                sv[t] = sv[t+off]; si[t] = si[t+off];
            }
        }
        __syncthreads();
    }
    int idx = si[0];
    zq[(size_t)row * DD + t] = cb[(size_t)idx * DD + t];   // t in [0,128)
}

// ---------------- graph mean pooling ----------------
__global__ __launch_bounds__(128) void graph_mean(
    const float* __restrict__ h, float* __restrict__ res)
{
    int b = blockIdx.x, d = threadIdx.x;
    float acc = 0.0f;
    const float* hp = h + (size_t)b * NPG * DD + d;
    for (int n = 0; n < NPG; ++n) acc += hp[n * DD];
    res[b * DD + d] = acc * (1.0f / NPG);
}

// ---------------- attention softmax + mix (tiny) ----------------
__global__ __launch_bounds__(256) void attn_mix(
    const float* __restrict__ q, const float* __restrict__ k,
    const float* __restrict__ v, float* __restrict__ attv)
{
    int t = threadIdx.x;               // one block of 256 = NB*HH
    int b = t >> 2, hh = t & 3;
    const int dh = DD / HH;            // 32
    const float* qp = q + b * DD + hh * dh;
    float sc[KK]; float mx = -3.4e38f;
#pragma unroll
    for (int kk = 0; kk < KK; ++kk) {
        const float* kp = k + (size_t)(b * KK + kk) * DD + hh * dh;
        float d = 0.0f;
#pragma unroll
        for (int i = 0; i < 32; ++i) d = fmaf(qp[i], kp[i], d);
        sc[kk] = d * 0.17677669529663687f;      // 1/sqrt(32)
        mx = fmaxf(mx, sc[kk]);
    }
    float s = 0.0f;
#pragma unroll
    for (int kk = 0; kk < KK; ++kk) { sc[kk] = __expf(sc[kk] - mx); s += sc[kk]; }
    float inv = 1.0f / s;
    float out[32];
#pragma unroll
    for (int i = 0; i < 32; ++i) out[i] = 0.0f;
#pragma unroll
    for (int kk = 0; kk < KK; ++kk) {
        float w = sc[kk] * inv;
        const float* vp = v + (size_t)(b * KK + kk) * DD + hh * dh;
#pragma unroll
        for (int i = 0; i < 32; ++i) out[i] = fmaf(w, vp[i], out[i]);
    }
#pragma unroll
    for (int i = 0; i < 32; ++i) attv[b * DD + hh * dh + i] = out[i];
}

// ---------------- concat [residue, att] ----------------
__global__ __launch_bounds__(256) void concat_kernel(
    const float* __restrict__ res, const float* __restrict__ att,
    float* __restrict__ cat)
{
    int idx = blockIdx.x * 256 + threadIdx.x;     // NB*2*DD threads
    int b = idx >> 8, j = idx & 255;
    cat[(size_t)b * 256 + j] = (j < DD) ? res[b * DD + j] : att[b * DD + (j - DD)];
}

// ---------------- gated fusion ----------------
__global__ __launch_bounds__(256) void fuse_kernel(
    const float* __restrict__ g, const float* __restrict__ res,
    const float* __restrict__ att, float* __restrict__ fused)
{
    int idx = blockIdx.x * 256 + threadIdx.x;     // NB*DD threads
    float gv = g[idx];
    fused[idx] = gv * res[idx] + (1.0f - gv) * att[idx];
}

// ---------------- final 256 -> 2 classifier head ----------------
__global__ __launch_bounds__(128) void cls_final(
    const float* __restrict__ z2, const float* __restrict__ W,
    const float* __restrict__ b, float* __restrict__ out)
{
    int t = threadIdx.x;               // 128 = NB*2
    int bb = t >> 1, o = t & 1;
    float acc = b[o];
    const float* r = z2 + bb * 256;
    for (int j = 0; j < 256; ++j) acc = fmaf(r[j], W[j * 2 + o], acc);
    out[bb * 2 + o] = acc;
}

// =====================================================================
extern "C" void kernel_launch(void* const* d_in, const int* in_sizes, int n_in,
                              void* d_out, int out_size, void* d_ws, size_t ws_size,
                              hipStream_t stream)
{
    const int*   x      = (const int*)d_in[0];
    const int*   ei     = (const int*)d_in[1];   // [2,E]: row0 src, row1 dst
    const float* emb    = (const float*)d_in[3];
    const float* gin_W1 = (const float*)d_in[4];
    const float* gin_b1 = (const float*)d_in[5];
    const float* gin_W2 = (const float*)d_in[6];
    const float* gin_b2 = (const float*)d_in[7];
    const float* bn_g   = (const float*)d_in[8];
    const float* bn_b   = (const float*)d_in[9];
    const float* pW1    = (const float*)d_in[10];
    const float* pb1    = (const float*)d_in[11];
    const float* pW2    = (const float*)d_in[12];
    const float* pb2    = (const float*)d_in[13];
    const float* Wq     = (const float*)d_in[14];
    const float* Wk     = (const float*)d_in[15];
    const float* Wv     = (const float*)d_in[16];
    const float* Wo     = (const float*)d_in[17];
    const float* gW1    = (const float*)d_in[18];
    const float* gb1    = (const float*)d_in[19];
    const float* gW2    = (const float*)d_in[20];
    const float* gb2    = (const float*)d_in[21];
    const float* cbk    = (const float*)d_in[22];
    const float* cW1    = (const float*)d_in[23];
    const float* cb1    = (const float*)d_in[24];
    const float* cW2    = (const float*)d_in[25];
    const float* cb2    = (const float*)d_in[26];
    const float* cW3    = (const float*)d_in[27];
    const float* cb3    = (const float*)d_in[28];
    float* out = (float*)d_out;

    // ---- workspace layout (fp32, 256B-aligned slices) ----
    float* W = (float*)d_ws;
    size_t o = 0;
    auto alloc = [&](size_t nf) { float* p = W + o; o += (nf + 63) & ~(size_t)63; return p; };
    float* h     = alloc((size_t)NN * DD);     // node features
    float* agg   = alloc((size_t)NN * DD);     // h + neighbor sum
    float* t1    = alloc((size_t)NN * 256);    // GIN hidden
    float* S     = alloc((size_t)NN * KK);
    float* cf    = alloc((size_t)NB * KK * DD);
    float* zq    = alloc((size_t)NB * KK * DD);
    float* res   = alloc((size_t)NB * DD);
    float* qb    = alloc((size_t)NB * DD);
    float* kb    = alloc((size_t)NB * KK * DD);
    float* vb    = alloc((size_t)NB * KK * DD);
    float* attv  = alloc((size_t)NB * DD);
    float* att   = alloc((size_t)NB * DD);
    float* cat   = alloc((size_t)NB * 256);
    float* g1b   = alloc((size_t)NB * 64);
    float* gb    = alloc((size_t)NB * DD);
    float* fused = alloc((size_t)NB * DD);
    float* c1b   = alloc((size_t)NB * 512);
    float* c2b   = alloc((size_t)NB * 256);

    const float bnscale = 0.99999500003749969f;  // 1/sqrt(1+1e-5)
    const dim3 blk(128);

    // ---- embedding ----
    embed_kernel<<<(NN * DD) / 256, 256, 0, stream>>>(x, emb, h);

    // ---- 3 GIN layers ----
    for (int l = 0; l < 3; ++l) {
        hipMemcpyAsync(agg, h, (size_t)NN * DD * sizeof(float),
                       hipMemcpyDeviceToDevice, stream);
        edge_scatter<<<(EE * 8) / 256, 256, 0, stream>>>(ei, ei + EE, h, agg);
        // [NN,128] @ [128,256] + b, relu
        wmma_gemm_f32<4, 256, 128><<<NN/16, blk, 0, stream>>>(
            agg, gin_W1 + (size_t)l * DD * 256, gin_b1 + l * 256,
            nullptr, nullptr, t1, ACT_RELU, 1.0f);
        // [NN,256] @ [256,128] + b, eval-BN, relu (layers 0,1)
        wmma_gemm_f32<2, 128, 256><<<NN/16, blk, 0, stream>>>(
            t1, gin_W2 + (size_t)l * 256 * DD, gin_b2 + l * DD,
            bn_g + l * DD, bn_b + l * DD, h,
            (l < 2) ? ACT_RELU : ACT_NONE, bnscale);
    }

    // ---- soft partitioner + cluster features + VQ ----
    partition_softmax<<<NN / 256, 256, 0, stream>>>(h, pW1, pb1, pW2, pb2, S);
    cluster_feat<<<NB, 256, 0, stream>>>(h, S, cf);
    vq_nearest<<<NB * KK, 128, 0, stream>>>(cf, cbk, zq);

    // ---- mean pooling ----
    graph_mean<<<NB, 128, 0, stream>>>(h, res);

    // ---- cluster attention ----
    wmma_gemm_f32<2, 128, 128><<<NB/16, blk, 0, stream>>>(
        res, Wq, nullptr, nullptr, nullptr, qb, ACT_NONE, 1.0f);
    wmma_gemm_f32<2, 128, 128><<<(NB*KK)/16, blk, 0, stream>>>(
        zq, Wk, nullptr, nullptr, nullptr, kb, ACT_NONE, 1.0f);
    wmma_gemm_f32<2, 128, 128><<<(NB*KK)/16, blk, 0, stream>>>(
        zq, Wv, nullptr, nullptr, nullptr, vb, ACT_NONE, 1.0f);
    attn_mix<<<1, 256, 0, stream>>>(qb, kb, vb, attv);
    wmma_gemm_f32<2, 128, 128><<<NB/16, blk, 0, stream>>>(
        attv, Wo, nullptr, nullptr, nullptr, att, ACT_NONE, 1.0f);

    // ---- gated fusion ----
    concat_kernel<<<(NB * 256) / 256, 256, 0, stream>>>(res, att, cat);
    wmma_gemm_f32<1, 64, 256><<<NB/16, blk, 0, stream>>>(
        cat, gW1, gb1, nullptr, nullptr, g1b, ACT_RELU, 1.0f);
    wmma_gemm_f32<2, 128, 64><<<NB/16, blk, 0, stream>>>(
        g1b, gW2, gb2, nullptr, nullptr, gb, ACT_SIG, 1.0f);
    fuse_kernel<<<(NB * DD) / 256, 256, 0, stream>>>(gb, res, att, fused);

    // ---- classifier ----
    wmma_gemm_f32<8, 512, 128><<<NB/16, blk, 0, stream>>>(
        fused, cW1, cb1, nullptr, nullptr, c1b, ACT_RELU, 1.0f);
    wmma_gemm_f32<4, 256, 512><<<NB/16, blk, 0, stream>>>(
        c1b, cW2, cb2, nullptr, nullptr, c2b, ACT_RELU, 1.0f);
    cls_final<<<1, 128, 0, stream>>>(c2b, cW3, cb3, out);
}